// VanillaGCN_63239098466423
// MI455X (gfx1250) — compile-verified
//
#include <hip/hip_runtime.h>

#define D_DIM 512
#define FIN_DIM 6
#define DEPTH_N 10
#define CHUNKS 1024

typedef __attribute__((ext_vector_type(2))) float v2f;
typedef __attribute__((ext_vector_type(8))) float v8f;

// ---------------- Deterministic CSR build (no atomics) ----------------

__global__ void k_zero_u16(unsigned short* __restrict__ p, int n) {
  int i = blockIdx.x * blockDim.x + threadIdx.x;
  if (i < n) p[i] = 0;
}

// One thread per edge-chunk: count dst occurrences into its private column.
__global__ void k_chunk_count(const int* __restrict__ dst,
                              unsigned short* __restrict__ LC, int E, int epc) {
  int c = blockIdx.x * blockDim.x + threadIdx.x;
  if (c >= CHUNKS) return;
  int s = c * epc;
  int e = s + epc; if (e > E) e = E;
  for (int i = s; i < e; ++i) {
    int d = dst[i];
    LC[(size_t)d * CHUNKS + c] += 1;   // column c owned exclusively by this thread
  }
}

// Per node: exclusive scan across chunks (in place) + total degree.
__global__ void k_col_scan(unsigned short* __restrict__ LC, int* __restrict__ deg, int n) {
  int v = blockIdx.x * blockDim.x + threadIdx.x;
  if (v >= n) return;
  unsigned short* row = LC + (size_t)v * CHUNKS;
  int run = 0;
  for (int c = 0; c < CHUNKS; ++c) {
    int t = row[c];
    row[c] = (unsigned short)run;
    run += t;
  }
  deg[v] = run;
}

// Single-block exclusive scan of degrees -> row_start[0..n].
__global__ void k_scan_deg(const int* __restrict__ deg, int* __restrict__ row_start, int n) {
  __shared__ int partial[256];
  __shared__ int base[256];
  int t = threadIdx.x;
  int chunk = (n + 255) / 256;
  int s = 0;
  for (int i = 0; i < chunk; ++i) { int idx = t * chunk + i; if (idx < n) s += deg[idx]; }
  partial[t] = s;
  __syncthreads();
  if (t == 0) { int run = 0; for (int i = 0; i < 256; ++i) { base[i] = run; run += partial[i]; } }
  __syncthreads();
  int run = base[t];
  for (int i = 0; i < chunk; ++i) {
    int idx = t * chunk + i;
    if (idx < n) { row_start[idx] = run; run += deg[idx]; }
  }
  if (t == 255) row_start[n] = run;
}

// Stable replay: slot = row_start[d] + chunk-prefix + running within-chunk rank.
__global__ void k_csr_fill(const int* __restrict__ src, const int* __restrict__ dst,
                           const int* __restrict__ row_start,
                           unsigned short* __restrict__ LC, int* __restrict__ col_idx,
                           int E, int epc) {
  int c = blockIdx.x * blockDim.x + threadIdx.x;
  if (c >= CHUNKS) return;
  int s = c * epc;
  int e = s + epc; if (e > E) e = E;
  for (int i = s; i < e; ++i) {
    int d = dst[i];
    size_t li = (size_t)d * CHUNKS + c;
    int off = LC[li];
    LC[li] = (unsigned short)(off + 1);
    col_idx[row_start[d] + off] = src[i];
  }
}

// ---------------- Input layer ----------------

__global__ void k_agg_fin(const float* __restrict__ feat, const int* __restrict__ row_start,
                          const int* __restrict__ col_idx, float* __restrict__ agg6, int n) {
  int v = blockIdx.x * blockDim.x + threadIdx.x;
  if (v >= n) return;
  float s0 = 0.f, s1 = 0.f, s2 = 0.f, s3 = 0.f, s4 = 0.f, s5 = 0.f;
  int b = row_start[v], e = row_start[v + 1];
  for (int i = b; i < e; ++i) {
    const float* f = feat + (size_t)col_idx[i] * FIN_DIM;
    s0 += f[0]; s1 += f[1]; s2 += f[2]; s3 += f[3]; s4 += f[4]; s5 += f[5];
  }
  float* o = agg6 + (size_t)v * FIN_DIM;
  o[0] = s0; o[1] = s1; o[2] = s2; o[3] = s3; o[4] = s4; o[5] = s5;
}

__global__ void k_input_layer(const float* __restrict__ agg6, const float* __restrict__ W_in,
                              const float* __restrict__ b_in, float* __restrict__ x, int n) {
  int idx = blockIdx.x * blockDim.x + threadIdx.x;
  if (idx >= n * D_DIM) return;
  int v = idx >> 9;
  int d = idx & (D_DIM - 1);
  const float* a = agg6 + (size_t)v * FIN_DIM;
  float s = b_in[d];
#pragma unroll
  for (int k = 0; k < FIN_DIM; ++k) s += a[k] * W_in[k * D_DIM + d];
  x[idx] = fmaxf(s, 0.0f);   // ReLU only after input layer (per reference)
}

// ---------------- Hidden-layer aggregation: one block per node ----------------

__global__ void __launch_bounds__(256)
k_aggregate(const float* __restrict__ x, const int* __restrict__ row_start,
            const int* __restrict__ col_idx, float* __restrict__ agg) {
  int v = blockIdx.x;
  int t = threadIdx.x;
  int b = row_start[v], e = row_start[v + 1];
  float a0 = 0.f, a1 = 0.f;
  for (int i = b; i < e; ++i) {
    const float* xr = x + (size_t)col_idx[i] * D_DIM;
    if (i + 1 < e)   // uniform condition: b,e are per-block scalars
      __builtin_prefetch(x + (size_t)col_idx[i + 1] * D_DIM + t, 0, 0);
    a0 += xr[t];
    a1 += xr[t + 256];
  }
  agg[(size_t)v * D_DIM + t] = a0;
  agg[(size_t)v * D_DIM + t + 256] = a1;
}

// ---------------- WMMA fp32 GEMM: Y[M,512] = A[M,512] @ W[512,512] + bias ----------------
// Workgroup: 8 waves -> 128(M) x 64(N) tile; wave w owns rows [16w,16w+16), 4 N-tiles.
// A staged row-major (pad 36), B staged TRANSPOSED (pad 34) so every WMMA fragment is one
// aligned 8-byte LDS load into a contiguous even VGPR pair (ds_load_b64, no repack movs).

#define GTM 128
#define GTN 64
#define GKC 32
#define APAD 36
#define BPAD 34

__global__ void __launch_bounds__(256)
k_gemm_bias_wmma(const float* __restrict__ A, const float* __restrict__ W,
                 const float* __restrict__ bias, float* __restrict__ Y, int M) {
  __shared__ float As[GTM][APAD];     // [row][k], stride 36: float4-aligned stores, b64 frag reads
  __shared__ float BsT[GTN][BPAD];    // [n][k],  stride 34: b64 frag reads, even alignment
  const int K = D_DIM, N = D_DIM;
  const int rBase = blockIdx.x * GTM;
  const int cBase = blockIdx.y * GTN;
  const int tid = threadIdx.x;
  const int lane = tid & 31, wave = tid >> 5;
  const int l16 = lane & 15, half = lane >> 4;

  v8f zero = {};
  v8f acc[4];
#pragma unroll
  for (int i = 0; i < 4; ++i) acc[i] = zero;

  for (int k0 = 0; k0 < K; k0 += GKC) {
    // Stage A tile (128x32): 4 x float4 per thread, coalesced b128 loads.
#pragma unroll
    for (int i = tid; i < GTM * GKC / 4; i += 256) {
      int r = i >> 3;             // 8 float4 per row
      int kk = (i & 7) * 4;
      int gr = rBase + r; if (gr >= M) gr = M - 1;
      const float4 v = *(const float4*)(A + (size_t)gr * K + k0 + kk);
      *(float4*)(&As[r][kk]) = v;
    }
    // Stage B tile (32x64) transposed: 2 x float4 per thread + 4-way scatter into BsT.
#pragma unroll
    for (int i = tid; i < GKC * GTN / 4; i += 256) {
      int r = i >> 4;             // 16 float4 per k-row
      int nn = (i & 15) * 4;
      const float4 v = *(const float4*)(W + (size_t)(k0 + r) * N + cBase + nn);
      BsT[nn + 0][r] = v.x;
      BsT[nn + 1][r] = v.y;
      BsT[nn + 2][r] = v.z;
      BsT[nn + 3][r] = v.w;
    }
    __syncthreads();

#pragma unroll
    for (int ks = 0; ks < GKC; ks += 4) {
      // A 16x4 frag: lanes 0-15 hold K+0/K+1, lanes 16-31 hold K+2/K+3 (krow even -> aligned)
      int krow = ks + 2 * half;
      v2f a = *(const v2f*)(&As[wave * 16 + l16][krow]);
#pragma unroll
      for (int nt = 0; nt < 4; ++nt) {
        // B 4x16 frag: same K-split across lane halves, N = l16; contiguous in BsT
        v2f b = *(const v2f*)(&BsT[nt * 16 + l16][krow]);
        acc[nt] = __builtin_amdgcn_wmma_f32_16x16x4_f32(
            false, a, false, b, (short)0, acc[nt], false, false);
      }
    }
    __syncthreads();
  }

  // C/D layout: VGPR i -> row i (lanes 0-15) or i+8 (lanes 16-31), col = l16.
#pragma unroll
  for (int nt = 0; nt < 4; ++nt) {
#pragma unroll
    for (int i = 0; i < 8; ++i) {
      int gr = rBase + wave * 16 + half * 8 + i;
      int gc = cBase + nt * 16 + l16;
      if (gr < M) Y[(size_t)gr * N + gc] = acc[nt][i] + bias[gc];
    }
  }
}

// ---------------- Output layer: dot(x[v,:], W_out) + b_out ----------------

__global__ void __launch_bounds__(256)
k_output(const float* __restrict__ x, const float* __restrict__ W_out,
         const float* __restrict__ b_out, float* __restrict__ out, int n) {
  int lane = threadIdx.x & 31;
  int wave = threadIdx.x >> 5;
  int v = blockIdx.x * 8 + wave;
  if (v >= n) return;
  const float* xr = x + (size_t)v * D_DIM;
  float p = 0.f;
#pragma unroll
  for (int j = 0; j < D_DIM / 32; ++j) {
    int d = lane + j * 32;
    p += xr[d] * W_out[d];
  }
#pragma unroll
  for (int off = 16; off >= 1; off >>= 1) p += __shfl_xor(p, off, 32);
  if (lane == 0) out[v] = p + b_out[0];
}

// ---------------- Host orchestration ----------------

extern "C" void kernel_launch(void* const* d_in, const int* in_sizes, int n_in,
                              void* d_out, int out_size, void* d_ws, size_t ws_size,
                              hipStream_t stream) {
  const float* features = (const float*)d_in[0];
  const float* W_in  = (const float*)d_in[1];
  const float* b_in  = (const float*)d_in[2];
  const float* Ws    = (const float*)d_in[3];
  const float* bs    = (const float*)d_in[4];
  const float* W_out = (const float*)d_in[5];
  const float* b_out = (const float*)d_in[6];
  const int*   src   = (const int*)d_in[7];
  const int*   dst   = (const int*)d_in[8];
  float* out = (float*)d_out;

  const int n = in_sizes[0] / FIN_DIM;            // 10000 nodes
  const int E = in_sizes[7];                      // 160000 edges
  const int epc = (E + CHUNKS - 1) / CHUNKS;

  // Carve workspace (~63 MB), 256B aligned.
  char* p = (char*)d_ws;
  auto carve = [&](size_t bytes) -> void* {
    void* r = (void*)p;
    p += (bytes + 255) & ~(size_t)255;
    return r;
  };
  float* xbuf    = (float*)carve((size_t)n * D_DIM * sizeof(float));
  float* aggbuf  = (float*)carve((size_t)n * D_DIM * sizeof(float));
  float* agg6    = (float*)carve((size_t)n * FIN_DIM * sizeof(float));
  unsigned short* LC = (unsigned short*)carve((size_t)n * CHUNKS * sizeof(unsigned short));
  int* deg       = (int*)carve((size_t)n * sizeof(int));
  int* row_start = (int*)carve((size_t)(n + 1) * sizeof(int));
  int* col_idx   = (int*)carve((size_t)E * sizeof(int));

  // 1) Deterministic CSR build.
  int lcElems = n * CHUNKS;
  k_zero_u16<<<(lcElems + 255) / 256, 256, 0, stream>>>(LC, lcElems);
  k_chunk_count<<<CHUNKS / 128, 128, 0, stream>>>(dst, LC, E, epc);
  k_col_scan<<<(n + 255) / 256, 256, 0, stream>>>(LC, deg, n);
  k_scan_deg<<<1, 256, 0, stream>>>(deg, row_start, n);
  k_csr_fill<<<CHUNKS / 128, 128, 0, stream>>>(src, dst, row_start, LC, col_idx, E, epc);

  // 2) Input layer: agg6 -> relu(agg6 @ W_in + b_in).
  k_agg_fin<<<(n + 255) / 256, 256, 0, stream>>>(features, row_start, col_idx, agg6, n);
  k_input_layer<<<(n * D_DIM + 255) / 256, 256, 0, stream>>>(agg6, W_in, b_in, xbuf, n);

  // 3) 10 hidden GCN layers: aggregate then WMMA GEMM (in-place ping through aggbuf).
  dim3 ggrid((n + GTM - 1) / GTM, D_DIM / GTN);
  for (int l = 0; l < DEPTH_N; ++l) {
    k_aggregate<<<n, 256, 0, stream>>>(xbuf, row_start, col_idx, aggbuf);
    k_gemm_bias_wmma<<<ggrid, 256, 0, stream>>>(aggbuf, Ws + (size_t)l * D_DIM * D_DIM,
                                                bs + (size_t)l * D_DIM, xbuf, n);
  }

  // 4) Output projection.
  k_output<<<(n + 7) / 8, 256, 0, stream>>>(xbuf, W_out, b_out, out, n);
}